// pLayer_88098369176005
// MI455X (gfx1250) — compile-verified
//
#include <hip/hip_runtime.h>
#include <cmath>

// ---------------------------------------------------------------------------
// pLayer forward (MI455X / gfx1250, wave32, WMMA fp32 path)
//
// Shapes: theta [M=1024, N=512], a_prev [B=256, 1022]. Power term is
// factorized so the whole forward is a single fused GEMM
//   C[256 x 1024] = A'[256 x 2048] @ Bm[2048 x 1024]
// with A' = [a_ext | a_neg], Bm = [[W*pos | g~*pos], [W*neg | g~*neg]],
// computed with V_WMMA_F32_16X16X4_F32, plus O(M*N) reductions.
//
// v2: LDS B tile stored transposed (Bs[col][k], stride 66) so each WMMA B
// fragment is a single aligned ds_load_b64 into an even VGPR pair (removes
// the v_dual_mov marshaling seen in v1); next B chunk prefetched with
// global_prefetch_b8 while the current chunk is being consumed.
// ---------------------------------------------------------------------------

typedef float v2f __attribute__((ext_vector_type(2)));
typedef float v8f __attribute__((ext_vector_type(8)));

#define GMIN_  0.01f
#define GMAX_  0.05f
#define ACT_E1 0.05f
#define ACT_E2 1.0f
#define ACT_E3 0.1f
#define ACT_E4 10.0f
#define NEG_E1 0.05f
#define NEG_E2 1.0f
#define NEG_E3 0.0f
#define NEG_E4 10.0f

constexpr int M_    = 1024;   // n_in + 2
constexpr int NPRE  = 1022;   // n_in
constexpr int NOUT  = 512;
constexpr int BATCH = 256;
constexpr int K2    = 2048;   // 2*M  (GEMM K)
constexpr int NC    = 1024;   // 2*NOUT (GEMM N: [z | y])
constexpr int LDSS  = 66;     // LDS row stride (even -> 8B-aligned pairs)

__device__ __forceinline__ float theta_eff(float t) {
    float c = fminf(fmaxf(t, -GMAX_), GMAX_);
    return (fabsf(c) < GMIN_) ? 0.0f : c;
}

// ---- per-column stats of theta_eff: sum|.|, max|.|, sum of g~ -------------
__global__ void colstats_kernel(const float* __restrict__ theta,
                                float* __restrict__ sumAbs,
                                float* __restrict__ maxG,
                                float* __restrict__ sumGt) {
    int n = blockIdx.x * blockDim.x + threadIdx.x;
    if (n >= NOUT) return;
    float s = 0.f, mx = 0.f;
    for (int m = 0; m < M_; ++m) {
        float g = fabsf(theta_eff(theta[m * NOUT + n]));
        s += g;
        mx = fmaxf(mx, g);
    }
    sumAbs[n] = s;
    maxG[n]   = mx;
    sumGt[n]  = (mx > 0.f) ? s * (GMAX_ / mx) : 0.f;
}

// ---- build fused B matrix  Bm[2048 x 1024] --------------------------------
__global__ void buildB_kernel(const float* __restrict__ theta,
                              const float* __restrict__ sumAbs,
                              const float* __restrict__ maxG,
                              float* __restrict__ Bm) {
    int i = blockIdx.x * blockDim.x + threadIdx.x;      // i < M_*NOUT
    int m = i >> 9, n = i & (NOUT - 1);
    float t  = theta_eff(theta[m * NOUT + n]);
    float g  = fabsf(t);
    float sA = sumAbs[n];
    float w  = (sA > 0.f) ? g / sA : 0.f;
    float mx = maxG[n];
    float gt = (mx > 0.f) ? g * (GMAX_ / mx) : 0.f;
    bool  pos = (t >= 0.f);
    Bm[(size_t)m * NC + n]                 = pos ? w  : 0.f;   // Wp
    Bm[(size_t)m * NC + NOUT + n]          = pos ? gt : 0.f;   // Gp
    Bm[(size_t)(M_ + m) * NC + n]          = pos ? 0.f : w;    // Wn
    Bm[(size_t)(M_ + m) * NC + NOUT + n]   = pos ? 0.f : gt;   // Gn
}

// ---- build A' = [a_ext | a_neg]  [256 x 2048] -----------------------------
__global__ void buildA_kernel(const float* __restrict__ aprev,
                              float* __restrict__ Ap) {
    int i = blockIdx.x * blockDim.x + threadIdx.x;      // i < BATCH*M_
    int b = i >> 10, m = i & (M_ - 1);
    float ae;
    if (m < NPRE)       ae = aprev[b * NPRE + m];
    else if (m == NPRE) ae = 1.0f;   // appended ones column
    else                ae = 0.0f;   // appended zeros column
    float an = (m == M_ - 1) ? 0.0f
             : -(NEG_E1 + NEG_E2 * tanhf((ae - NEG_E3) * NEG_E4));
    Ap[(size_t)b * K2 + m]      = ae;
    Ap[(size_t)b * K2 + M_ + m] = an;
}

// ---- batch-summed squares per input row (for rank-1 power term) -----------
__global__ void rowstats_kernel(const float* __restrict__ Ap,
                                float* __restrict__ sa2,
                                float* __restrict__ sn2) {
    int m = blockIdx.x * blockDim.x + threadIdx.x;
    if (m >= M_) return;
    float s1 = 0.f, s2 = 0.f;
    for (int b = 0; b < BATCH; ++b) {
        float ae = Ap[(size_t)b * K2 + m];
        float an = Ap[(size_t)b * K2 + M_ + m];
        s1 += ae * ae;
        s2 += an * an;
    }
    sa2[m] = s1;
    sn2[m] = s2;
}

// ---- fused GEMM: C[256x1024] = A'[256x2048] @ Bm[2048x1024] ---------------
// 4 waves / block; block tile 64x64; wave tile 16x64 (4 col sub-tiles).
// B chunks [64k x 64n] staged transposed in LDS; fp32 WMMA 16x16x4.
__global__ __launch_bounds__(128)
void gemm_wmma_kernel(const float* __restrict__ A,
                      const float* __restrict__ Bm,
                      float* __restrict__ C) {
    __shared__ float Bs[64 * LDSS];              // Bs[col*LDSS + k], ~16.9 KB
    const int lane = threadIdx.x & 31;
    const int wave = threadIdx.x >> 5;
    const int half = lane >> 4;                  // lane group 0/1 (K split)
    const int idx  = lane & 15;                  // M (A) / N (B,C) position
    const int rowBase = blockIdx.y * 64 + wave * 16;
    const int colBase = blockIdx.x * 64;
    const int arow = rowBase + idx;

    v8f acc0 = {}, acc1 = {}, acc2 = {}, acc3 = {};

    for (int kc = 0; kc < K2; kc += 64) {
        __syncthreads();
        // cooperative load of B chunk, stored transposed: Bs[cc][kr].
        // Global side stays coalesced (consecutive tids -> consecutive cc);
        // LDS side: lanes hit banks (2*cc + kr) % 64, stride 2 -> no conflict.
        for (int i = threadIdx.x; i < 64 * 64; i += 128) {
            int kr = i >> 6, cc = i & 63;
            Bs[cc * LDSS + kr] = Bm[(size_t)(kc + kr) * NC + colBase + cc];
        }
        // prefetch next chunk into cache while this one is consumed
        if (kc + 64 < K2) {
            for (int i = threadIdx.x; i < 64 * 64; i += 128 * 8) {
                int kr = i >> 6, cc = i & 63;
                __builtin_prefetch(&Bm[(size_t)(kc + 64 + kr) * NC + colBase + cc], 0, 1);
            }
        }
        __syncthreads();
        #pragma unroll 4
        for (int kk = 0; kk < 64; kk += 4) {
            const int k0 = kk + 2 * half;        // ISA A/B frag layout:
            // lanes0-15 hold K=kk,kk+1 ; lanes16-31 hold K=kk+2,kk+3
            v2f a  = *(const v2f*)(A + (size_t)arow * K2 + kc + k0);
            v2f b0 = *(const v2f*)&Bs[( 0 + idx) * LDSS + k0];
            v2f b1 = *(const v2f*)&Bs[(16 + idx) * LDSS + k0];
            v2f b2 = *(const v2f*)&Bs[(32 + idx) * LDSS + k0];
            v2f b3 = *(const v2f*)&Bs[(48 + idx) * LDSS + k0];
            acc0 = __builtin_amdgcn_wmma_f32_16x16x4_f32(false, a, false, b0,
                       (short)0, acc0, false, false);
            acc1 = __builtin_amdgcn_wmma_f32_16x16x4_f32(false, a, false, b1,
                       (short)0, acc1, false, false);
            acc2 = __builtin_amdgcn_wmma_f32_16x16x4_f32(false, a, false, b2,
                       (short)0, acc2, false, false);
            acc3 = __builtin_amdgcn_wmma_f32_16x16x4_f32(false, a, false, b3,
                       (short)0, acc3, false, false);
        }
    }
    // C/D layout: VGPR v -> row (v + 8*half), col = idx (+16 per sub-tile)
    #pragma unroll
    for (int v = 0; v < 8; ++v) {
        int row = rowBase + v + 8 * half;
        float* crow = C + (size_t)row * NC + colBase + idx;
        crow[0]  = acc0[v];
        crow[16] = acc1[v];
        crow[32] = acc2[v];
        crow[48] = acc3[v];
    }
}

// ---- reductions for power -------------------------------------------------
__global__ void zero_kernel(float* accum) { *accum = 0.f; }

__device__ __forceinline__ void block_reduce_add(float v, float* accum) {
    __shared__ float red[256];
    red[threadIdx.x] = v;
    __syncthreads();
    for (int s = 128; s > 0; s >>= 1) {
        if ((int)threadIdx.x < s) red[threadIdx.x] += red[threadIdx.x + s];
        __syncthreads();
    }
    if (threadIdx.x == 0) atomicAdd(accum, red[0]);
}

// term1 = sum_{m,n} Gp*sa2[m] + Gn*sn2[m]
__global__ void term1_kernel(const float* __restrict__ Bm,
                             const float* __restrict__ sa2,
                             const float* __restrict__ sn2,
                             float* accum) {
    int i = blockIdx.x * blockDim.x + threadIdx.x;      // i < M_*NOUT
    int m = i >> 9, n = i & (NOUT - 1);
    float v = Bm[(size_t)m * NC + NOUT + n] * sa2[m]
            + Bm[(size_t)(M_ + m) * NC + NOUT + n] * sn2[m];
    block_reduce_add(v, accum);
}

// term2+term3 = sum_{b,n} z*(z*sumGt[n] - 2*y)
__global__ void term23_kernel(const float* __restrict__ C,
                              const float* __restrict__ sumGt,
                              float* accum) {
    int i = blockIdx.x * blockDim.x + threadIdx.x;      // i < BATCH*NOUT
    int b = i >> 9, n = i & (NOUT - 1);
    float z = C[(size_t)b * NC + n];
    float y = C[(size_t)b * NC + NOUT + n];
    float v = z * (z * sumGt[n] - 2.0f * y);
    block_reduce_add(v, accum);
}

// ---- activation + power write --------------------------------------------
__global__ void act_kernel(const float* __restrict__ C,
                           float* __restrict__ out) {
    int i = blockIdx.x * blockDim.x + threadIdx.x;      // i < BATCH*NOUT
    int b = i >> 9, n = i & (NOUT - 1);
    float z = C[(size_t)b * NC + n];
    out[i] = ACT_E1 + ACT_E2 * tanhf((z - ACT_E3) * ACT_E4);
}

__global__ void finalize_kernel(const float* accum, float* __restrict__ out) {
    out[BATCH * NOUT] = accum[0] * ((float)BATCH / (float)(BATCH - 1));
}

// ---------------------------------------------------------------------------
extern "C" void kernel_launch(void* const* d_in, const int* in_sizes, int n_in,
                              void* d_out, int out_size, void* d_ws, size_t ws_size,
                              hipStream_t stream) {
    const float* a_prev = (const float*)d_in[0];   // [256, 1022]
    const float* theta  = (const float*)d_in[1];   // [1024, 512]
    float* out = (float*)d_out;                    // 256*512 + 1 floats

    // workspace layout (floats), ~11.6 MB total
    float* ws     = (float*)d_ws;
    float* Ap     = ws;                              // 256*2048
    float* Bm     = Ap + (size_t)BATCH * K2;         // 2048*1024
    float* Cm     = Bm + (size_t)K2 * NC;            // 256*1024
    float* sumAbs = Cm + (size_t)BATCH * NC;         // 512
    float* maxG   = sumAbs + NOUT;                   // 512
    float* sumGt  = maxG + NOUT;                     // 512
    float* sa2    = sumGt + NOUT;                    // 1024
    float* sn2    = sa2 + M_;                        // 1024
    float* accum  = sn2 + M_;                        // 1

    (void)in_sizes; (void)n_in; (void)out_size; (void)ws_size;

    colstats_kernel<<<dim3(2), dim3(256), 0, stream>>>(theta, sumAbs, maxG, sumGt);
    buildB_kernel<<<dim3((M_ * NOUT) / 256), dim3(256), 0, stream>>>(theta, sumAbs, maxG, Bm);
    buildA_kernel<<<dim3((BATCH * M_) / 256), dim3(256), 0, stream>>>(a_prev, Ap);
    rowstats_kernel<<<dim3(M_ / 256), dim3(256), 0, stream>>>(Ap, sa2, sn2);
    zero_kernel<<<dim3(1), dim3(1), 0, stream>>>(accum);

    gemm_wmma_kernel<<<dim3(NC / 64, BATCH / 64), dim3(128), 0, stream>>>(Ap, Bm, Cm);

    term1_kernel<<<dim3((M_ * NOUT) / 256), dim3(256), 0, stream>>>(Bm, sa2, sn2, accum);
    term23_kernel<<<dim3((BATCH * NOUT) / 256), dim3(256), 0, stream>>>(Cm, sumGt, accum);
    act_kernel<<<dim3((BATCH * NOUT) / 256), dim3(256), 0, stream>>>(Cm, out);
    finalize_kernel<<<dim3(1), dim3(1), 0, stream>>>(accum, out);
}